// ConvAttention_8461085573456
// MI455X (gfx1250) — compile-verified
//
#include <hip/hip_runtime.h>
#include <math.h>

typedef __attribute__((ext_vector_type(2))) float v2f;
typedef __attribute__((ext_vector_type(8))) float v8f;
typedef int b128_t __attribute__((vector_size(4 * sizeof(int))));

#if __has_builtin(__builtin_amdgcn_global_load_async_to_lds_b128) && \
    __has_builtin(__builtin_amdgcn_s_wait_asynccnt)
#define HAS_ASYNC_LDS 1
typedef __attribute__((address_space(1))) b128_t* gptr_b128;
typedef __attribute__((address_space(3))) b128_t* lptr_b128;
#else
#define HAS_ASYNC_LDS 0
#endif

__device__ __forceinline__ v8f wmma_f32_16x16x4(v2f a, v2f b, v8f c) {
  // 8 args: (neg_a, A, neg_b, B, c_mod, C, reuse_a, reuse_b)
  return __builtin_amdgcn_wmma_f32_16x16x4_f32(false, a, false, b, (short)0, c,
                                               false, false);
}

// ---------------------------------------------------------------------------
// Conv main loop, templated on GUARD so interior tiles (the common case) get
// unconditional B loads with no per-lane select / exec masking.
// ---------------------------------------------------------------------------
template <int TAPS, int NB, bool GUARD>
__device__ __forceinline__ void conv_mainloop(const float* __restrict__ W,
                                              const float* __restrict__ X,
                                              v8f (&acc)[NB], int K, int N,
                                              int am, int n0, int lidx,
                                              int half, int b0, int pad) {
  for (int tap = 0; tap < TAPS; ++tap) {
    const int tcol = n0 + lidx + tap - pad;  // shifted input column
    const bool colok = !GUARD || ((tcol >= 0) && (tcol < N));
    for (int k0 = 0; k0 < K; k0 += 4) {
      const int ka = k0 + 2 * half;
      v2f a;
      if (TAPS == 1) {
        a = *(const v2f*)(W + (size_t)am * K + ka);  // contiguous: one b64
      } else {
        a.x = W[(size_t)(am * K + ka) * TAPS + tap];
        a.y = W[(size_t)(am * K + ka + 1) * TAPS + tap];
      }
      v2f bf[NB];
#pragma unroll
      for (int bb = 0; bb < NB; ++bb) {
        const float* Xb = X + (size_t)(b0 + bb) * K * N;
        bf[bb].x = colok ? Xb[(size_t)ka * N + tcol] : 0.f;
        bf[bb].y = colok ? Xb[(size_t)(ka + 1) * N + tcol] : 0.f;
      }
#pragma unroll
      for (int bb = 0; bb < NB; ++bb)
        acc[bb] = wmma_f32_16x16x4(a, bf[bb], acc[bb]);
    }
  }
}

// ---------------------------------------------------------------------------
// Generic conv1d-as-GEMM: Y[b][m][n] = act( sum_{k,tap} W[m][k][tap] *
//                                           X[b][k][n+tap-pad] + bias[m] )
// One wave -> one 16x16 (M x N) tile for NB consecutive batches. The A
// (weight) fragment is loaded once per k-step and feeds NB back-to-back
// WMMAs. Requires: M % 16 == 0, K % 4 == 0 (true for all convs here).
// ---------------------------------------------------------------------------
template <int TAPS, bool RELU, int NB>
__global__ void conv_wmma(const float* __restrict__ W,
                          const float* __restrict__ bias,
                          const float* __restrict__ X, float* __restrict__ Y,
                          int M, int N, int K, int nbatch) {
  const int lane = threadIdx.x & 31;
  const int half = lane >> 4;   // selects K rows {0,1} vs {2,3} of the frag
  const int lidx = lane & 15;   // M row (A) / N col (B, C)
  const int wv = blockIdx.x * (blockDim.x >> 5) + (threadIdx.x >> 5);

  const int mtiles = M >> 4;
  const int ntiles = (N + 15) >> 4;
  const int bgroups = nbatch / NB;
  const int total = mtiles * ntiles * bgroups;
  if (wv >= total) return;  // wave-uniform: EXEC stays all-ones for WMMA

  int t = wv;
  const int bg = t / (mtiles * ntiles);
  t -= bg * (mtiles * ntiles);
  const int mt = t / ntiles;
  const int nt = t - mt * ntiles;
  const int m0 = mt << 4, n0 = nt << 4, b0 = bg * NB;

  const v8f zero = {0.f, 0.f, 0.f, 0.f, 0.f, 0.f, 0.f, 0.f};
  v8f acc[NB];
#pragma unroll
  for (int i = 0; i < NB; ++i) acc[i] = zero;

  const int pad = (TAPS - 1) / 2;
  const int am = m0 + lidx;  // A-matrix row for this lane (both halves)

  // wave-uniform: does this tile touch any out-of-range input column?
  const bool interior = (n0 - pad >= 0) && (n0 + 15 + (TAPS - 1) - pad < N);
  if (interior) {
    conv_mainloop<TAPS, NB, false>(W, X, acc, K, N, am, n0, lidx, half, b0, pad);
  } else {
    conv_mainloop<TAPS, NB, true>(W, X, acc, K, N, am, n0, lidx, half, b0, pad);
  }

  const int n = n0 + lidx;
#pragma unroll
  for (int r = 0; r < 8; ++r) {
    const int m = m0 + r + 8 * half;  // C/D layout: VGPR r, upper half +8
    const float bv = bias[m];
    if (n < N) {
#pragma unroll
      for (int bb = 0; bb < NB; ++bb) {
        float v = acc[bb][r] + bv;
        if (RELU) v = fmaxf(v, 0.f);
        Y[(size_t)(b0 + bb) * M * N + (size_t)m * N + n] = v;
      }
    }
  }
}

// ---------------------------------------------------------------------------
// ||q||^2 over channels (B=16, C=80, T1=800) and ||k||^2 (T2=200)
// ---------------------------------------------------------------------------
__global__ void norms_kernel(const float* __restrict__ q,
                             const float* __restrict__ k,
                             float* __restrict__ qn, float* __restrict__ kn) {
  const int i = blockIdx.x * blockDim.x + threadIdx.x;
  const int NQ = 16 * 800;
  if (i < NQ) {
    const int b = i / 800, t = i - b * 800;
    const float* p = q + (size_t)b * 80 * 800 + t;
    float s = 0.f;
#pragma unroll
    for (int c = 0; c < 80; ++c) { float v = p[(size_t)c * 800]; s += v * v; }
    qn[i] = s;
  } else if (i < NQ + 16 * 200) {
    const int j = i - NQ;
    const int b = j / 200, t = j - b * 200;
    const float* p = k + (size_t)b * 80 * 200 + t;
    float s = 0.f;
#pragma unroll
    for (int c = 0; c < 80; ++c) { float v = p[(size_t)c * 200]; s += v * v; }
    kn[j] = s;
  }
}

// ---------------------------------------------------------------------------
// Fused attention. Block = 5 waves = one batch's 50 M-tiles / 10 blocks.
// The k matrix (80x200 f32) is staged through LDS in 20-channel chunks using
// CDNA5 async global->LDS b128 copies (ASYNCcnt), double-buffered so the next
// chunk's DMA overlaps the current chunk's 65 WMMAs. LDS rows are padded to
// 208 floats with a zeroed tail so all 13 B-fragment loads are unconditional
// (no exec-masked DS loads in the hot loop). Each wave owns rows [m0,m0+16)
// of T1 and all 13 N-tiles of T2=200: QK via WMMA f32 16x16x4 over 80
// channels, then -5e-4*(q2+k2-2qk), in-register row log-softmax over T2
// (xor-shuffles 1..8 stay inside each 16-lane half, matching the C-matrix
// layout), + log(prior + 1e-8).
// ---------------------------------------------------------------------------
#define CHUNK_C 20                  // channels per staged chunk (4 chunks)
#define LDS_W 208                   // padded LDS row stride (200 + 8 zeros)
#define CHUNK_F (CHUNK_C * 200)     // real floats per chunk

__device__ __forceinline__ void stage_chunk(const float* __restrict__ g,
                                            float* l) {
  const int tid = threadIdx.x;  // 0..159
#pragma unroll
  for (int i = 0; i < 7; ++i) {
    const int idx = tid + i * 160;  // 1000 float4 elements (20 rows x 50)
    if (idx < CHUNK_F / 4) {
      const int row = idx / 50;
      const int col = idx - row * 50;
      const float* src = g + (size_t)row * 200 + col * 4;
      float* dst = l + (size_t)row * LDS_W + col * 4;
#if HAS_ASYNC_LDS
      __builtin_amdgcn_global_load_async_to_lds_b128((gptr_b128)src,
                                                     (lptr_b128)dst, 0, 0);
#else
      *(float4*)dst = *(const float4*)src;
#endif
    }
  }
}

__device__ __forceinline__ void wait_stage() {
#if HAS_ASYNC_LDS
  __builtin_amdgcn_s_wait_asynccnt(0);
#endif
  __syncthreads();
}

__global__ __launch_bounds__(160) void attn_kernel(
    const float* __restrict__ q, const float* __restrict__ k,
    const float* __restrict__ qn, const float* __restrict__ kn,
    const float* __restrict__ prior, float* __restrict__ out) {
  __shared__ float kbuf[2][CHUNK_C * LDS_W];  // ~33 KB double buffer

  const int lane = threadIdx.x & 31;
  const int half = lane >> 4;
  const int lidx = lane & 15;
  const int wave = threadIdx.x >> 5;            // 0..4
  const int b = blockIdx.y;                     // batch
  const int mt = blockIdx.x * 5 + wave;         // 0..49 exactly
  const int m0 = mt << 4;
  const float* qb = q + (size_t)b * 80 * 800;
  const float* kb = k + (size_t)b * 80 * 200;

  // zero the 8-column pad of every LDS row once (both buffers)
  for (int i = threadIdx.x; i < 2 * CHUNK_C * 8; i += 160) {
    const int buf = i / (CHUNK_C * 8);
    const int rem = i - buf * (CHUNK_C * 8);
    const int row = rem >> 3, cp = rem & 7;
    kbuf[buf][row * LDS_W + 200 + cp] = 0.f;
  }

  const v8f zero = {0.f, 0.f, 0.f, 0.f, 0.f, 0.f, 0.f, 0.f};
  v8f acc[13];
#pragma unroll
  for (int t = 0; t < 13; ++t) acc[t] = zero;

  stage_chunk(kb, kbuf[0]);  // prologue: chunk 0 in flight

  for (int c = 0; c < 4; ++c) {
    wait_stage();  // chunk c resident + pads zeroed + visible to all waves
    if (c + 1 < 4) stage_chunk(kb + (c + 1) * CHUNK_F, kbuf[(c + 1) & 1]);
    const float* kl = kbuf[c & 1];

    for (int k0 = 0; k0 < CHUNK_C; k0 += 4) {
      const int kg = c * CHUNK_C + k0 + 2 * half;  // global channel
      const int kl0 = k0 + 2 * half;               // channel within chunk
      v2f a;
      a.x = qb[(size_t)kg * 800 + m0 + lidx];
      a.y = qb[(size_t)(kg + 1) * 800 + m0 + lidx];
      v2f bf[13];
#pragma unroll
      for (int t = 0; t < 13; ++t) {
        const int n = t * 16 + lidx;  // n<208 always valid (zero pad)
        bf[t].x = kl[kl0 * LDS_W + n];
        bf[t].y = kl[(kl0 + 1) * LDS_W + n];
      }
#pragma unroll
      for (int t = 0; t < 13; ++t) acc[t] = wmma_f32_16x16x4(a, bf[t], acc[t]);
    }
  }

  const float NEG_INF = -__builtin_inff();
  float knv[13];
#pragma unroll
  for (int t = 0; t < 13; ++t) {
    const int n = t * 16 + lidx;
    knv[t] = (n < 200) ? kn[b * 200 + n] : 0.f;
  }
  float qnv[8];
#pragma unroll
  for (int r = 0; r < 8; ++r) qnv[r] = qn[b * 800 + m0 + r + 8 * half];

  // logits in-place; out-of-range T2 columns -> -inf (excluded from softmax)
#pragma unroll
  for (int t = 0; t < 13; ++t) {
    const bool ok = (t * 16 + lidx) < 200;
#pragma unroll
    for (int r = 0; r < 8; ++r) {
      const float l = -0.0005f * (qnv[r] + knv[t] - 2.0f * acc[t][r]);
      acc[t][r] = ok ? l : NEG_INF;
    }
  }

  // row-wise logsumexp across 13 tiles then across the 16 lanes of this half
  float lse[8];
#pragma unroll
  for (int r = 0; r < 8; ++r) {
    float m = NEG_INF;
#pragma unroll
    for (int t = 0; t < 13; ++t) m = fmaxf(m, acc[t][r]);
#pragma unroll
    for (int o = 1; o < 16; o <<= 1) m = fmaxf(m, __shfl_xor(m, o, 32));
    float s = 0.f;
#pragma unroll
    for (int t = 0; t < 13; ++t) s += expf(acc[t][r] - m);
#pragma unroll
    for (int o = 1; o < 16; o <<= 1) s += __shfl_xor(s, o, 32);
    lse[r] = m + logf(s);
  }

#pragma unroll
  for (int t = 0; t < 13; ++t) {
    const int n = t * 16 + lidx;
    if (n < 200) {
#pragma unroll
      for (int r = 0; r < 8; ++r) {
        const int m = m0 + r + 8 * half;
        const size_t idx = (size_t)b * 160000 + (size_t)m * 200 + n;
        out[idx] = acc[t][r] - lse[r] + logf(prior[idx] + 1e-8f);
      }
    }
  }
}

// ---------------------------------------------------------------------------
extern "C" void kernel_launch(void* const* d_in, const int* in_sizes, int n_in,
                              void* d_out, int out_size, void* d_ws,
                              size_t ws_size, hipStream_t stream) {
  (void)in_sizes; (void)n_in; (void)out_size; (void)ws_size;
  const float* queries = (const float*)d_in[0];   // (16,80,800)
  const float* keys    = (const float*)d_in[1];   // (16,256,200)
  const float* prior   = (const float*)d_in[2];   // (16,800,200)
  const float* kw1 = (const float*)d_in[3];       // (512,256,3)
  const float* kb1 = (const float*)d_in[4];
  const float* kw2 = (const float*)d_in[5];       // (80,512,1)
  const float* kb2 = (const float*)d_in[6];
  const float* qw1 = (const float*)d_in[7];       // (160,80,3)
  const float* qb1 = (const float*)d_in[8];
  const float* qw2 = (const float*)d_in[9];       // (80,160,1)
  const float* qb2 = (const float*)d_in[10];
  const float* qw3 = (const float*)d_in[11];      // (80,80,1)
  const float* qb3 = (const float*)d_in[12];
  float* out = (float*)d_out;

  // scratch layout (floats); total ~6.01M floats (~24 MB), L2-resident
  float* ws = (float*)d_ws;
  float* k1 = ws;                       // 16*512*200
  float* kk = k1 + 16 * 512 * 200;      // 16*80*200
  float* q1 = kk + 16 * 80 * 200;       // 16*160*800
  float* q2 = q1 + 16 * 160 * 800;      // 16*80*800
  float* qq = q2 + 16 * 80 * 800;       // 16*80*800
  float* qn = qq + 16 * 80 * 800;       // 16*800
  float* kn = qn + 16 * 800;            // 16*200

  auto nblk = [](int M, int N, int bgroups) {
    const int tiles = (M / 16) * ((N + 15) / 16) * bgroups;
    return (tiles + 7) / 8;  // 8 waves per 256-thread block
  };

  // key encoder
  conv_wmma<3, true, 8><<<nblk(512, 200, 2), 256, 0, stream>>>(
      kw1, kb1, keys, k1, 512, 200, 256, 16);
  conv_wmma<1, false, 8><<<nblk(80, 200, 2), 256, 0, stream>>>(
      kw2, kb2, k1, kk, 80, 200, 512, 16);
  // query encoder
  conv_wmma<3, true, 8><<<nblk(160, 800, 2), 256, 0, stream>>>(
      qw1, qb1, queries, q1, 160, 800, 80, 16);
  conv_wmma<1, true, 8><<<nblk(80, 800, 2), 256, 0, stream>>>(
      qw2, qb2, q1, q2, 80, 800, 160, 16);
  conv_wmma<1, false, 8><<<nblk(80, 800, 2), 256, 0, stream>>>(
      qw3, qb3, q2, qq, 80, 800, 80, 16);
  // norms + fused attention/log-softmax
  norms_kernel<<<(16 * (800 + 200) + 255) / 256, 256, 0, stream>>>(qq, kk, qn, kn);
  attn_kernel<<<dim3(10, 16), 160, 0, stream>>>(qq, kk, qn, kn, prior, out);
}